// MPNNNet_5214090297997
// MI455X (gfx1250) — compile-verified
//
#include <hip/hip_runtime.h>
#include <math.h>

typedef __attribute__((ext_vector_type(2))) float v2f;
typedef __attribute__((ext_vector_type(8))) float v8f;

#define NFEAT  128
#define NHID   64
#define NCLASS 40

// ---------------------------------------------------------------------------
// GEMM1: H[N,64] = X[N,128] @ W1^T + b1   (W1 is [64,128] row-major)
// One wave computes a 16(M) x 64(N) tile via 4 accumulators of
// v_wmma_f32_16x16x4_f32, K-loop of 128 in steps of 4.
// A frag layout (16x4 f32): lane<16 -> K=k,k+1 ; lane>=16 -> K=k+2,k+3
// B frag layout (4x16 f32): VGPR0={K=k,K=k+2}, VGPR1={K=k+1,K=k+3} -> float2
//   per lane = W1[n][kk], W1[n][kk+1] with kk = k + (lane>>4)*2, n = tile*16 + (lane&15)
// ---------------------------------------------------------------------------
__global__ __launch_bounds__(256) void mpnn_gemm1_wmma(
    const float* __restrict__ X, const float* __restrict__ W1,
    const float* __restrict__ b1, float* __restrict__ H, int nRows)
{
    const int wave = (blockIdx.x * blockDim.x + threadIdx.x) >> 5;
    const int lane = threadIdx.x & 31;
    const int row0 = wave * 16;
    if (row0 + 16 > nRows) return;          // nRows divisible by 16 (100000)

    const int lmod  = lane & 15;
    const int lhalf = lane >> 4;            // selects K pair

    const float* __restrict__ xrow = X  + (size_t)(row0 + lmod) * NFEAT + lhalf * 2;
    const float* __restrict__ wrow = W1 + (size_t)lmod * NFEAT + lhalf * 2;

    v8f acc[4] = {v8f{}, v8f{}, v8f{}, v8f{}};

#pragma unroll 4
    for (int k = 0; k < NFEAT; k += 4) {
        v2f a = *(const v2f*)(xrow + k);
#pragma unroll
        for (int t = 0; t < 4; ++t) {
            v2f b = *(const v2f*)(wrow + (size_t)t * 16 * NFEAT + k);
            acc[t] = __builtin_amdgcn_wmma_f32_16x16x4_f32(
                false, a, false, b, (short)0, acc[t], false, false);
        }
    }

    // C layout: VGPR j -> lanes 0..15 = (M=j, N=lmod), lanes 16..31 = (M=j+8, N=lmod)
#pragma unroll
    for (int t = 0; t < 4; ++t) {
        const int   n    = t * 16 + lmod;
        const float bias = b1[n];
#pragma unroll
        for (int j = 0; j < 8; ++j) {
            const int r = row0 + j + lhalf * 8;
            H[(size_t)r * NHID + n] = acc[t][j] + bias;
        }
    }
}

// ---------------------------------------------------------------------------
// GEMM2: H2[N,40] = Hagg[N,64] @ W2^T + b2   (W2 is [40,64] row-major)
// Same scheme, 3 column tiles (48 padded, cols >= 40 masked on store).
// ---------------------------------------------------------------------------
__global__ __launch_bounds__(256) void mpnn_gemm2_wmma(
    const float* __restrict__ Hagg, const float* __restrict__ W2,
    const float* __restrict__ b2, float* __restrict__ H2, int nRows)
{
    const int wave = (blockIdx.x * blockDim.x + threadIdx.x) >> 5;
    const int lane = threadIdx.x & 31;
    const int row0 = wave * 16;
    if (row0 + 16 > nRows) return;

    const int lmod  = lane & 15;
    const int lhalf = lane >> 4;

    const float* __restrict__ xrow = Hagg + (size_t)(row0 + lmod) * NHID + lhalf * 2;

    v8f acc[3] = {v8f{}, v8f{}, v8f{}};

#pragma unroll 4
    for (int k = 0; k < NHID; k += 4) {
        v2f a = *(const v2f*)(xrow + k);
#pragma unroll
        for (int t = 0; t < 3; ++t) {
            const int n  = t * 16 + lmod;
            const int nc = (n < NCLASS) ? n : 0;   // pad lanes read row 0, never stored
            v2f b = *(const v2f*)(W2 + (size_t)nc * NHID + lhalf * 2 + k);
            acc[t] = __builtin_amdgcn_wmma_f32_16x16x4_f32(
                false, a, false, b, (short)0, acc[t], false, false);
        }
    }

#pragma unroll
    for (int t = 0; t < 3; ++t) {
        const int n = t * 16 + lmod;
        if (n < NCLASS) {
            const float bias = b2[n];
#pragma unroll
            for (int j = 0; j < 8; ++j) {
                const int r = row0 + j + lhalf * 8;
                H2[(size_t)r * NCLASS + n] = acc[t][j] + bias;
            }
        }
    }
}

// ---------------------------------------------------------------------------
// Edge aggregation: one wave per edge (E real edges + N self loops).
// Lanes stride the feature dim (coalesced 128B rows); scatter via
// global_atomic_add_f32 — working set is L2-resident (192 MB L2).
// ---------------------------------------------------------------------------
__global__ __launch_bounds__(256) void mpnn_aggregate(
    const float* __restrict__ Hin, const int* __restrict__ eidx,
    float* __restrict__ Sum, float* __restrict__ Cnt,
    int nEdges, int nNodes, int nFeat, int doCnt)
{
    const long long wid = ((long long)blockIdx.x * blockDim.x + threadIdx.x) >> 5;
    const int lane = threadIdx.x & 31;
    const long long total = (long long)nEdges + nNodes;
    if (wid >= total) return;

    // prefetch the index stream ahead (speculative, OOB-safe)
    __builtin_prefetch(&eidx[wid + 2048], 0, 0);

    int s, d;
    if (wid < nEdges) {
        s = eidx[wid];
        d = eidx[(size_t)nEdges + wid];
    } else {
        s = d = (int)(wid - nEdges);      // self loop
    }

    const float* __restrict__ hs = Hin + (size_t)s * nFeat;
    float*       __restrict__ sd = Sum + (size_t)d * nFeat;
    for (int f = lane; f < nFeat; f += 32)
        atomicAdd(&sd[f], hs[f]);
    if (doCnt && lane == 0)
        atomicAdd(&Cnt[d], 1.0f);
}

// ---------------------------------------------------------------------------
// Finalize layer 1: Hagg = relu(Sum / Cnt), in place over Sum.  (relu twice == once)
// ---------------------------------------------------------------------------
__global__ __launch_bounds__(256) void mpnn_finalize1(
    float* __restrict__ Sum, const float* __restrict__ Cnt, int nNodes)
{
    const long long total = (long long)nNodes * NHID;
    for (long long t = (long long)blockIdx.x * blockDim.x + threadIdx.x;
         t < total; t += (long long)gridDim.x * blockDim.x) {
        const long long i = t >> 6;                 // / NHID
        const float v = Sum[t] * (1.0f / Cnt[i]);
        Sum[t] = fmaxf(v, 0.0f);
    }
}

// ---------------------------------------------------------------------------
// Finalize layer 2 + log_softmax: one wave per node over 40 classes
// (lane covers class f = lane and lane+32). wave32 shuffles.
// ---------------------------------------------------------------------------
__global__ __launch_bounds__(256) void mpnn_logsoftmax(
    const float* __restrict__ Sum2, const float* __restrict__ Cnt,
    float* __restrict__ Out, int nNodes)
{
    const int wid  = (blockIdx.x * blockDim.x + threadIdx.x) >> 5;
    const int lane = threadIdx.x & 31;
    if (wid >= nNodes) return;

    const float inv = 1.0f / Cnt[wid];
    const float* __restrict__ s = Sum2 + (size_t)wid * NCLASS;

    const bool has2 = (lane + 32) < NCLASS;         // lanes 0..7
    const float v0 = s[lane] * inv;
    const float v1 = has2 ? s[lane + 32] * inv : -INFINITY;

    float m = fmaxf(v0, v1);
#pragma unroll
    for (int off = 16; off > 0; off >>= 1)
        m = fmaxf(m, __shfl_xor(m, off, 32));

    float se = expf(v0 - m) + (has2 ? expf(v1 - m) : 0.0f);
#pragma unroll
    for (int off = 16; off > 0; off >>= 1)
        se += __shfl_xor(se, off, 32);

    const float lse = m + logf(se);
    Out[(size_t)wid * NCLASS + lane] = v0 - lse;
    if (has2)
        Out[(size_t)wid * NCLASS + lane + 32] = v1 - lse;
}

// ---------------------------------------------------------------------------
// Workspace layout (floats):
//   [0,            N*64)  : H1, later reused for H2 (N*40 <= N*64)
//   [N*64,       2*N*64)  : Sum1 -> Hagg (in place) -> Sum2 (after gemm2 reads Hagg)
//   [2*N*64, 2*N*64 + N)  : Cnt
// ---------------------------------------------------------------------------
extern "C" void kernel_launch(void* const* d_in, const int* in_sizes, int n_in,
                              void* d_out, int out_size, void* d_ws, size_t ws_size,
                              hipStream_t stream)
{
    const float* x    = (const float*)d_in[0];
    const int*   eidx = (const int*)  d_in[1];
    const float* W1   = (const float*)d_in[2];
    const float* b1   = (const float*)d_in[3];
    const float* W2   = (const float*)d_in[4];
    const float* b2   = (const float*)d_in[5];

    const int nNodes = in_sizes[0] / NFEAT;   // 100000
    const int nEdges = in_sizes[1] / 2;       // 1600000

    float* H  = (float*)d_ws;                           // N*64
    float* S  = H + (size_t)nNodes * NHID;              // N*64
    float* C  = S + (size_t)nNodes * NHID;              // N

    // zero Sum1 + Cnt (contiguous)
    hipMemsetAsync(S, 0, ((size_t)nNodes * NHID + nNodes) * sizeof(float), stream);

    const int rowTiles   = (nNodes + 15) / 16;                  // 6250
    const int gemmBlocks = (rowTiles + 7) / 8;                  // 8 waves / block

    mpnn_gemm1_wmma<<<gemmBlocks, 256, 0, stream>>>(x, W1, b1, H, nNodes);

    const long long totalE   = (long long)nEdges + nNodes;
    const int       aggBlocks = (int)((totalE * 32 + 255) / 256);

    mpnn_aggregate<<<aggBlocks, 256, 0, stream>>>(H, eidx, S, C, nEdges, nNodes, NHID, 1);

    const long long elems1 = (long long)nNodes * NHID;
    mpnn_finalize1<<<(int)((elems1 + 255) / 256), 256, 0, stream>>>(S, C, nNodes);

    // H2 overwrites H1 slot; reads Hagg from S
    mpnn_gemm2_wmma<<<gemmBlocks, 256, 0, stream>>>(S, W2, b2, H, nNodes);

    // Sum2 reuses S (gemm2 has fully consumed Hagg by stream order)
    hipMemsetAsync(S, 0, (size_t)nNodes * NCLASS * sizeof(float), stream);

    mpnn_aggregate<<<aggBlocks, 256, 0, stream>>>(H, eidx, S, C, nEdges, nNodes, NCLASS, 0);

    mpnn_logsoftmax<<<(nNodes + 7) / 8, 256, 0, stream>>>(S, C, (float*)d_out, nNodes);
}